// GCN_8315056685500
// MI455X (gfx1250) — compile-verified
//
#include <hip/hip_runtime.h>
#include <hip/hip_bf16.h>

typedef float v2f __attribute__((ext_vector_type(2)));
typedef float v8f __attribute__((ext_vector_type(8)));

// ---------------------------------------------------------------------------
// GEMM: Hout[N,64] = X[N,64] @ W[64,64] using V_WMMA_F32_16X16X4_F32.
// Block = 128 threads = 4 waves; each wave computes a 16x64 tile.
// W is staged in LDS PRE-SWIZZLED into WMMA B-fragment order so each fragment
// is one ds_load_b64 (no VGPR shuffle movs between WMMAs):
//   Wf[kk][t][lane] = { W[4kk+2hi][16t+l16], W[4kk+2hi+1][16t+l16] }
// EXEC kept all-ones (WMMA requirement): OOB rows clamp loads, guard stores.
// ---------------------------------------------------------------------------
__global__ __launch_bounds__(128) void gemm64_wmma(
    const float* __restrict__ X, const float* __restrict__ W,
    float* __restrict__ Hout, int N) {
  __shared__ v2f Wf[16 * 4 * 32];   // [kstep][ntile][lane], 16KB
  const int tid = threadIdx.x;

  // fill fragment-ordered LDS image of W (once per block, L2-resident source)
  for (int idx = tid; idx < 2048; idx += 128) {
    const int lane2 = idx & 31;
    const int t     = (idx >> 5) & 3;
    const int kk    = idx >> 7;
    const int row   = 4 * kk + 2 * (lane2 >> 4);
    const int col   = 16 * t + (lane2 & 15);
    v2f w;
    w.x = W[row * 64 + col];
    w.y = W[(row + 1) * 64 + col];
    Wf[idx] = w;
  }
  __syncthreads();

  const int lane = tid & 31;
  const int l16  = lane & 15;
  const int hi   = lane >> 4;            // 0: K pair {k,k+1}; 1: {k+2,k+3}
  const int rowBase = blockIdx.x * 64 + (tid >> 5) * 16;
  int r = rowBase + l16;
  if (r > N - 1) r = N - 1;              // clamp load address, keep EXEC full

  v8f acc0 = {}, acc1 = {}, acc2 = {}, acc3 = {};
  const float* xrow = X + (size_t)r * 64;

#pragma unroll
  for (int kk = 0; kk < 16; ++kk) {
    // A fragment: 16x4 f32 (VGPR0=K0/K2, VGPR1=K1/K3 per half-wave)
    v2f a;
    a.x = xrow[4 * kk + 2 * hi];
    a.y = xrow[4 * kk + 2 * hi + 1];
    // B fragments: single b64 LDS load each, immediate offsets
    const v2f* fb = Wf + kk * 128 + lane;
    const v2f b0 = fb[0];
    const v2f b1 = fb[32];
    const v2f b2 = fb[64];
    const v2f b3 = fb[96];

    acc0 = __builtin_amdgcn_wmma_f32_16x16x4_f32(false, a, false, b0, (short)0, acc0, false, false);
    acc1 = __builtin_amdgcn_wmma_f32_16x16x4_f32(false, a, false, b1, (short)0, acc1, false, false);
    acc2 = __builtin_amdgcn_wmma_f32_16x16x4_f32(false, a, false, b2, (short)0, acc2, false, false);
    acc3 = __builtin_amdgcn_wmma_f32_16x16x4_f32(false, a, false, b3, (short)0, acc3, false, false);
  }

  // C/D layout: VGPR j holds M = rowBase + j + 8*hi, N = l16 (+16/32/48)
#pragma unroll
  for (int j = 0; j < 8; ++j) {
    const int m = rowBase + j + 8 * hi;
    if (m < N) {
      float* o = Hout + (size_t)m * 64 + l16;
      o[0]  = acc0[j];
      o[16] = acc1[j];
      o[32] = acc2[j];
      o[48] = acc3[j];
    }
  }
}

// ---------------------------------------------------------------------------
// Elementwise / scatter kernels
// ---------------------------------------------------------------------------
__global__ void zero_f32(float* __restrict__ p, int n) {
  int i = blockIdx.x * blockDim.x + threadIdx.x;
  if (i < n) p[i] = 0.0f;
}

__global__ void deg_kernel(const long long* __restrict__ ei, float* __restrict__ deg, int E) {
  int e = blockIdx.x * blockDim.x + threadIdx.x;
  if (e < E) atomicAdd(&deg[(int)ei[(size_t)E + e]], 1.0f);
}

__global__ void dis_kernel(const float* __restrict__ deg,
                           float* __restrict__ disI, float* __restrict__ disP, int N) {
  int i = blockIdx.x * blockDim.x + threadIdx.x;
  if (i < N) {
    float d = deg[i];
    disI[i] = rsqrtf(d + 2.0f);
    disP[i] = rsqrtf(d + 1.0f);
  }
}

// agg[i,f] = fill * disI[i]^2 * h[i,f]   (writes every element; no memset needed)
__global__ void init_agg(const float* __restrict__ h, const float* __restrict__ dis,
                         float* __restrict__ agg, int total, float fill) {
  int i = blockIdx.x * blockDim.x + threadIdx.x;
  if (i < total) {
    float d = dis[i >> 6];
    agg[i] = fill * d * d * h[i];
  }
}

// One wave32 per edge: each lane moves 2 features (float2 load + 2 atomic f32 adds).
__global__ __launch_bounds__(256) void scatter64(
    const long long* __restrict__ ei, const float* __restrict__ h,
    const float* __restrict__ dis, float* __restrict__ agg, int E) {
  const int lane = threadIdx.x & 31;
  const int e = (int)((blockIdx.x * 256u + threadIdx.x) >> 5);
  if (e >= E) return;                       // wave-uniform
  const int s = (int)ei[e];
  const int d = (int)ei[(size_t)E + e];
  const float norm = dis[s] * dis[d];
  const float2 v = ((const float2*)(h + (size_t)s * 64))[lane];
  float* ad = agg + (size_t)d * 64 + 2 * lane;
  atomicAdd(ad,     norm * v.x);
  atomicAdd(ad + 1, norm * v.y);
}

__global__ void bias_relu(const float* __restrict__ agg, const float* __restrict__ b,
                          float* __restrict__ out, int total) {
  int i = blockIdx.x * blockDim.x + threadIdx.x;
  if (i < total) {
    float v = agg[i] + b[i & 63];
    out[i] = v > 0.0f ? v : 0.0f;
  }
}

// z[i] = h[i,:] . Wout[:,0]
__global__ void gemv_out(const float* __restrict__ h, const float* __restrict__ Wout,
                         float* __restrict__ z, int N) {
  int i = blockIdx.x * blockDim.x + threadIdx.x;
  if (i >= N) return;
  const float* row = h + (size_t)i * 64;
  float acc = 0.0f;
#pragma unroll
  for (int f = 0; f < 64; ++f) acc += row[f] * Wout[f];
  z[i] = acc;
}

__global__ void init_out(const float* __restrict__ z, const float* __restrict__ disP,
                         const float* __restrict__ bout, float* __restrict__ out, int N) {
  int i = blockIdx.x * blockDim.x + threadIdx.x;
  if (i < N) {
    float d = disP[i];
    out[i] = bout[0] + d * d * z[i];
  }
}

__global__ void scatter_out(const long long* __restrict__ ei, const float* __restrict__ z,
                            const float* __restrict__ disP, float* __restrict__ out, int E) {
  int e = blockIdx.x * blockDim.x + threadIdx.x;
  if (e < E) {
    const int s = (int)ei[e];
    const int d = (int)ei[(size_t)E + e];
    atomicAdd(&out[d], disP[s] * disP[d] * z[s]);
  }
}

// ---------------------------------------------------------------------------
extern "C" void kernel_launch(void* const* d_in, const int* in_sizes, int n_in,
                              void* d_out, int out_size, void* d_ws, size_t ws_size,
                              hipStream_t stream) {
  const float*     x    = (const float*)d_in[0];
  const long long* ei   = (const long long*)d_in[1];   // int64 [2,E] flat
  const float*     Win  = (const float*)d_in[2];
  const float*     bin  = (const float*)d_in[3];
  const float*     Wh1  = (const float*)d_in[4];
  const float*     bh1  = (const float*)d_in[5];
  const float*     Wh2  = (const float*)d_in[6];
  const float*     bh2  = (const float*)d_in[7];
  const float*     Wout = (const float*)d_in[8];
  const float*     bout = (const float*)d_in[9];
  float* out = (float*)d_out;

  const int N = in_sizes[0] / 64;
  const int E = in_sizes[1] / 2;
  const int NF = N * 64;

  // workspace layout (floats)
  float* bufA = (float*)d_ws;
  float* bufB = bufA + (size_t)NF;
  float* z    = bufB + (size_t)NF;
  float* disI = z + N;
  float* disP = disI + N;
  float* deg  = disP + N;

  const int T = 256;
  const unsigned gN   = (N + T - 1) / T;
  const unsigned gE   = (E + T - 1) / T;
  const unsigned gNF  = (NF + T - 1) / T;
  const unsigned gGem = (unsigned)((N + 63) / 64);
  const unsigned gSc  = (unsigned)(((size_t)E * 32 + T - 1) / T);

  // degree + normalization factors
  zero_f32<<<gN, T, 0, stream>>>(deg, N);
  deg_kernel<<<gE, T, 0, stream>>>(ei, deg, E);
  dis_kernel<<<gN, T, 0, stream>>>(deg, disI, disP, N);

  // layer 1: x -> A
  gemm64_wmma<<<gGem, 128, 0, stream>>>(x, Win, bufA, N);
  init_agg<<<gNF, T, 0, stream>>>(bufA, disI, bufB, NF, 2.0f);
  scatter64<<<gSc, T, 0, stream>>>(ei, bufA, disI, bufB, E);
  bias_relu<<<gNF, T, 0, stream>>>(bufB, bin, bufA, NF);

  // layer 2: A -> B
  gemm64_wmma<<<gGem, 128, 0, stream>>>(bufA, Wh1, bufB, N);
  init_agg<<<gNF, T, 0, stream>>>(bufB, disI, bufA, NF, 2.0f);
  scatter64<<<gSc, T, 0, stream>>>(ei, bufB, disI, bufA, E);
  bias_relu<<<gNF, T, 0, stream>>>(bufA, bh1, bufB, NF);

  // layer 3: B -> A
  gemm64_wmma<<<gGem, 128, 0, stream>>>(bufB, Wh2, bufA, N);
  init_agg<<<gNF, T, 0, stream>>>(bufA, disI, bufB, NF, 2.0f);
  scatter64<<<gSc, T, 0, stream>>>(ei, bufA, disI, bufB, E);
  bias_relu<<<gNF, T, 0, stream>>>(bufB, bh2, bufA, NF);

  // output layer (improved=False, fill=1): gemv + scalar scatter
  gemv_out<<<gN, T, 0, stream>>>(bufA, Wout, z, N);
  init_out<<<gN, T, 0, stream>>>(z, disP, bout, out, N);
  scatter_out<<<gE, T, 0, stream>>>(ei, z, disP, out, E);
}